// SpatialEmbLoss_3d_87187836109472
// MI455X (gfx1250) — compile-verified
//
#include <hip/hip_runtime.h>

// ---------------------------------------------------------------------------
// SpatialEmbLoss 3D for MI455X (gfx1250, wave32).
//   B=2, D=32, H=192, W=192, N_SIGMA=3, K=8
// Pipeline:
//   kInit      : zero accumulator block
//   kStats     : per-voxel masked reductions -> per-(b,k) stats + bg seed loss
//   kCoeff     : per-instance centers / s_exp / var_loss; builds 8x16 B-matrix
//   kDist      : V_WMMA_F32_16X16X4_F32 x2 computes d2 for 16 voxels x 8
//                instances per wave; dist=exp(-d2); emits sortable keys
//                (descending error, mask bit in mantissa LSB) + fg seed loss
//   radix sort : 4 passes x (kHist, kScanHist, kScatter) per (b,k); stable
//                8-bit LSD, 263KB dynamic LDS rank matrix (CDNA5 320KB WGP LDS)
//   kPosCount/kScanPos/kLovasz : Lovasz hinge via global positive-prefix
//   kFinal     : combine -> d_out[0]
// ---------------------------------------------------------------------------

typedef __attribute__((ext_vector_type(2))) float v2f;
typedef __attribute__((ext_vector_type(8))) float v8f;

#define Bv 2
#define Dv 32
#define Hv 192
#define Wv 192
#define Nv (Dv*Hv*Wv)            // 1179648
#define Kv 8

#define SORT_T 256
#define SORT_I 16
#define SORT_ELEMS (SORT_T*SORT_I)   // 4096
#define NBLK (Nv/SORT_ELEMS)         // 288
#define MAT_STRIDE 257               // LDS bank-conflict padding

// accumulator layout (float offsets inside ws[0..2047])
#define OFF_STAT  0     // [b][k][14]: cnt, sumXYZ*3, cmCnt, cmXYZ*3, mSig*3, mSig2*3
#define OFF_BG    224   // [b] bg seed loss
#define OFF_FG    226   // [b] fg seed loss
#define OFF_INST  228   // [b][k] lovasz per-instance loss
#define OFF_VAR   244   // [b] var_loss (already /obj)
#define OFF_VALID 246   // [b][k] valid flag
#define OFF_COEFF 272   // [8 rows][16 cols] WMMA B-matrix (per current batch)
#define ACC_FLOATS 512

static_assert(Nv % SORT_ELEMS == 0, "sort tiling");
static_assert((Nv/16) % 8 == 0, "wmma tiling");

__global__ void kInit(float* acc) { acc[threadIdx.x] = 0.0f; }

// ---------------------------------------------------------------------------
__global__ void kStats(const float* __restrict__ pred, const int* __restrict__ inst,
                       const int* __restrict__ lab, const unsigned char* __restrict__ cim,
                       float* __restrict__ acc)
{
  const int b = blockIdx.y;
  const int tid = threadIdx.x;
  __shared__ float a[113];
  if (tid < 113) a[tid] = 0.0f;
  __syncthreads();
  const size_t pb = (size_t)b * 7 * Nv;
  float bg = 0.0f;
  for (int i = 0; i < 8; ++i) {
    int v = blockIdx.x * 2048 + i * 256 + tid;
    int x = v % Wv; int r = v / Wv; int y = r % Hv; int z = r / Hv;
    float xm = x * (1.0f/(Wv-1)), ym = y * (1.0f/(Hv-1)), zm = z * (1.0f/(Dv-1));
    int iv = inst[(size_t)b*Nv + v];
    int lv = lab[(size_t)b*Nv + v];
    float sseed = 1.0f / (1.0f + expf(-pred[pb + (size_t)6*Nv + v]));
    if (lv == 0) bg += sseed * sseed;
    if (iv >= 1 && iv <= Kv) {
      float* base = &a[(iv-1) * 14];
      atomicAdd(&base[0], 1.0f);
      atomicAdd(&base[1], xm); atomicAdd(&base[2], ym); atomicAdd(&base[3], zm);
      float s0 = pred[pb + (size_t)3*Nv + v];
      float s1 = pred[pb + (size_t)4*Nv + v];
      float s2 = pred[pb + (size_t)5*Nv + v];
      atomicAdd(&base[8],  s0);    atomicAdd(&base[9],  s1);    atomicAdd(&base[10], s2);
      atomicAdd(&base[11], s0*s0); atomicAdd(&base[12], s1*s1); atomicAdd(&base[13], s2*s2);
      if (cim[(size_t)b*Nv + v]) {
        atomicAdd(&base[4], 1.0f);
        atomicAdd(&base[5], xm); atomicAdd(&base[6], ym); atomicAdd(&base[7], zm);
      }
    }
  }
  atomicAdd(&a[112], bg);
  __syncthreads();
  if (tid < 112) {
    int k = tid / 14, f = tid % 14;
    atomicAdd(&acc[(b*Kv + k)*14 + f], a[tid]);
  } else if (tid == 112) {
    atomicAdd(&acc[OFF_BG + b], a[112]);
  }
}

// ---------------------------------------------------------------------------
__global__ void kCoeff(float* __restrict__ acc, int b)
{
  __shared__ float ce[Kv][3], se[Kv][3], vr[Kv], vd[Kv];
  int t = threadIdx.x;
  if (t < Kv) {
    const float* s = &acc[(b*Kv + t) * 14];
    float cnt = s[0], safe = fmaxf(cnt, 1.0f);
    float cmc = s[4];
    float var = 0.0f;
    for (int c = 0; c < 3; ++c) {
      float mean = s[1+c] / safe;
      ce[t][c] = (cmc == 1.0f) ? s[5+c] : mean;   // cm_center is raw sum; count==1
      float ms = s[8+c], ms2 = s[11+c];
      float sm = ms / safe;                        // s_mean (== stop-grad copy)
      var += ms2 - 2.0f*sm*ms + sm*sm*cnt;
      se[t][c] = expf(10.0f * sm);
    }
    vr[t] = var / (3.0f * safe);
    vd[t] = (cnt > 0.0f) ? 1.0f : 0.0f;
  }
  __syncthreads();
  if (t < 128) {
    int r = t >> 4, n = t & 15;
    float val = 0.0f;
    if (n < Kv) {
      if (r < 3)       val = se[n][r];
      else if (r == 3) val = se[n][0]*ce[n][0]*ce[n][0] + se[n][1]*ce[n][1]*ce[n][1]
                           + se[n][2]*ce[n][2]*ce[n][2];
      else if (r < 7)  { int c = r - 4; val = -2.0f * se[n][c] * ce[n][c]; }
    }
    acc[OFF_COEFF + r*16 + n] = val;
  }
  if (t == 0) {
    float vs = 0.0f, v = 0.0f;
    for (int k = 0; k < Kv; ++k) { vs += vd[k]; v += vd[k]*vr[k]; }
    float obj = fmaxf(vs, 1.0f);
    acc[OFF_VAR + b] = v / obj;
    for (int k = 0; k < Kv; ++k) acc[OFF_VALID + b*Kv + k] = vd[k];
  }
}

// ---------------------------------------------------------------------------
// d2 = A(16x8 feat) @ Bcoef(8x16 inst) via two chained V_WMMA_F32_16X16X4_F32.
// A layout (ISA 16x4 f32): lane m=l&15; VGPR0 = K(0|2), VGPR1 = K(1|3) by l>=16.
// B layout (4x16): VGPR0 = rows(0|2), VGPR1 = rows(1|3) striped over lanes.
// C layout (16x16 f32): VGPR g -> row g (lanes 0-15) / g+8 (lanes 16-31), col=l&15.
__global__ void kDist(const float* __restrict__ pred, const int* __restrict__ inst,
                      float* __restrict__ acc, unsigned* __restrict__ keys, int b)
{
  const int tid  = threadIdx.x;
  const int lane = tid & 31, wave = tid >> 5;
  const int tile = blockIdx.x * 8 + wave;
  const int kh = lane >> 4, n = lane & 15;
  const float* cf = &acc[OFF_COEFF];
  const size_t pb = (size_t)b * 7 * Nv;

  // A fragment: this lane's voxel row
  int mv = lane & 15;
  int va = tile * 16 + mv;
  int x = va % Wv; int r2 = va / Wv; int y = r2 % Hv; int z = r2 / Hv;
  float e0 = tanhf(pred[pb + (size_t)0*Nv + va]) + x * (1.0f/(Wv-1));
  float e1 = tanhf(pred[pb + (size_t)1*Nv + va]) + y * (1.0f/(Hv-1));
  float e2 = tanhf(pred[pb + (size_t)2*Nv + va]) + z * (1.0f/(Dv-1));
  float f0 = e0*e0, f1 = e1*e1, f2 = e2*e2, f3 = 1.0f;
  float f4 = e0,    f5 = e1,    f6 = e2,    f7 = 0.0f;

  v2f a1, a2, bb1, bb2;
  a1.x = kh ? f2 : f0;  a1.y = kh ? f3 : f1;
  a2.x = kh ? f6 : f4;  a2.y = kh ? f7 : f5;
  bb1.x = cf[(2*kh + 0)*16 + n];      bb1.y = cf[(2*kh + 1)*16 + n];
  bb2.x = cf[(4 + 2*kh + 0)*16 + n];  bb2.y = cf[(4 + 2*kh + 1)*16 + n];

  v8f c = {0.f,0.f,0.f,0.f,0.f,0.f,0.f,0.f};
  c = __builtin_amdgcn_wmma_f32_16x16x4_f32(false, a1, false, bb1, (short)0, c, false, false);
  c = __builtin_amdgcn_wmma_f32_16x16x4_f32(false, a2, false, bb2, (short)0, c, false, false);

  float fg = 0.0f;
#pragma unroll
  for (int g = 0; g < 8; ++g) {
    int m = g + 8*kh;
    int v = tile * 16 + m;
    if (n < Kv) {
      float d2   = c[g];
      float dist = expf(-d2);
      int iv = inst[(size_t)b*Nv + v];
      int mk = (iv == n + 1) ? 1 : 0;
      float e = mk ? (2.0f - 2.0f*dist) : (2.0f*dist);   // 1 - (2d-1)*(2m-1)
      // descending-sortable key with label bit packed in mantissa LSB
      unsigned kb = (__float_as_uint(e) & ~1u) | (unsigned)mk;
      keys[(size_t)n*Nv + v] = ~kb;
      float sd = 1.0f / (1.0f + expf(-pred[pb + (size_t)6*Nv + v]));
      if (mk) { float dd = sd - dist; fg += dd*dd; }
    }
  }
  __shared__ float red[256];
  red[tid] = fg; __syncthreads();
  for (int off = 128; off; off >>= 1) { if (tid < off) red[tid] += red[tid+off]; __syncthreads(); }
  if (tid == 0) atomicAdd(&acc[OFF_FG + b], red[0]);
}

// ---------------------------------------------------------------------------
__global__ void kHist(const unsigned* __restrict__ in, unsigned* __restrict__ hist, int shift)
{
  __shared__ unsigned h[256];
  int tid = threadIdx.x, blk = blockIdx.x;
  h[tid] = 0; __syncthreads();
  for (int i = 0; i < SORT_I; ++i) {
    unsigned k = in[(size_t)blk*SORT_ELEMS + i*256 + tid];
    atomicAdd(&h[(k >> shift) & 255u], 1u);
  }
  __syncthreads();
  hist[(size_t)tid * NBLK + blk] = h[tid];
}

__global__ void kScanHist(unsigned* __restrict__ hist)
{
  __shared__ unsigned s[256];
  __shared__ unsigned carry;
  int tid = threadIdx.x;
  if (tid == 0) carry = 0;
  __syncthreads();
  for (int c = 0; c < NBLK; ++c) {
    unsigned v = hist[(size_t)c*256 + tid];
    s[tid] = v; __syncthreads();
    for (int off = 1; off < 256; off <<= 1) {
      unsigned t = (tid >= off) ? s[tid - off] : 0u;
      __syncthreads();
      s[tid] += t;
      __syncthreads();
    }
    unsigned incl = s[tid];
    hist[(size_t)c*256 + tid] = carry + incl - v;    // exclusive
    __syncthreads();
    if (tid == 255) carry += incl;
    __syncthreads();
  }
}

// stable 8-bit scatter; dynamic LDS rank matrix [256 digits][256 threads] (pad 257)
__global__ void kScatter(const unsigned* __restrict__ in, unsigned* __restrict__ out,
                         const unsigned* __restrict__ hist, int shift)
{
  extern __shared__ unsigned mat[];
  int tid = threadIdx.x, blk = blockIdx.x;
  for (int d = 0; d < 256; ++d) mat[d*MAT_STRIDE + tid] = 0;
  __syncthreads();
  unsigned it[SORT_I];
  size_t base = (size_t)blk*SORT_ELEMS + (size_t)tid*SORT_I;
#pragma unroll
  for (int j = 0; j < SORT_I; ++j) it[j] = in[base + j];
#pragma unroll
  for (int j = 0; j < SORT_I; ++j) mat[(((it[j] >> shift) & 255u))*MAT_STRIDE + tid]++;
  __syncthreads();
  { // column scan: thread owns digit d = tid
    unsigned run = hist[(size_t)tid * NBLK + blk];
    for (int j = 0; j < 256; ++j) {
      unsigned t = mat[tid*MAT_STRIDE + j];
      mat[tid*MAT_STRIDE + j] = run;
      run += t;
    }
  }
  __syncthreads();
#pragma unroll
  for (int j = 0; j < SORT_I; ++j) {
    unsigned d = (it[j] >> shift) & 255u;
    unsigned p = mat[d*MAT_STRIDE + tid];
    mat[d*MAT_STRIDE + tid] = p + 1;
    out[p] = it[j];
  }
}

// ---------------------------------------------------------------------------
__global__ void kPosCount(const unsigned* __restrict__ keys, unsigned* __restrict__ pos)
{
  __shared__ unsigned r[256];
  int tid = threadIdx.x, blk = blockIdx.x;
  unsigned c = 0;
  for (int i = 0; i < SORT_I; ++i)
    c += (~keys[(size_t)blk*SORT_ELEMS + i*256 + tid]) & 1u;
  r[tid] = c; __syncthreads();
  for (int off = 128; off; off >>= 1) { if (tid < off) r[tid] += r[tid+off]; __syncthreads(); }
  if (tid == 0) pos[blk] = r[0];
}

__global__ void kScanPos(unsigned* __restrict__ pos)
{
  unsigned run = 0;
  for (int i = 0; i < NBLK; ++i) { unsigned t = pos[i]; pos[i] = run; run += t; }
  pos[NBLK] = run;   // total positives P
}

__global__ void kLovasz(const unsigned* __restrict__ keys, const unsigned* __restrict__ pos,
                        float* __restrict__ acc, int slot)
{
  __shared__ unsigned su[256];
  __shared__ float   sf[256];
  int tid = threadIdx.x, blk = blockIdx.x;
  unsigned kk[SORT_I];
  size_t base = (size_t)blk*SORT_ELEMS + (size_t)tid*SORT_I;
  unsigned cnt = 0;
#pragma unroll
  for (int j = 0; j < SORT_I; ++j) { kk[j] = keys[base + j]; cnt += (~kk[j]) & 1u; }
  su[tid] = cnt; __syncthreads();
  for (int off = 1; off < 256; off <<= 1) {
    unsigned t = (tid >= off) ? su[tid - off] : 0u;
    __syncthreads();
    su[tid] += t;
    __syncthreads();
  }
  unsigned excl = su[tid] - cnt;
  float P = (float)pos[NBLK];
  unsigned p = pos[blk] + excl;
  float csum = 0.0f;
#pragma unroll
  for (int j = 0; j < SORT_I; ++j) {
    unsigned kb = ~kk[j];
    unsigned mk = kb & 1u;
    float e = __uint_as_float(kb);
    float pprev = (float)p;
    p += mk;
    float pf = (float)p;
    float gi = (float)(blk*SORT_ELEMS + tid*SORT_I + j + 1);   // 1-based rank
    float j1 = 1.0f - (P - pf) / (P + gi - pf);
    float j0 = (gi == 1.0f) ? 0.0f
                            : 1.0f - (P - pprev) / (P + (gi - 1.0f) - pprev);
    csum += fmaxf(e, 0.0f) * (j1 - j0);
  }
  sf[tid] = csum; __syncthreads();
  for (int off = 128; off; off >>= 1) { if (tid < off) sf[tid] += sf[tid+off]; __syncthreads(); }
  if (tid == 0) atomicAdd(&acc[OFF_INST + slot], sf[0]);
}

// ---------------------------------------------------------------------------
__global__ void kFinal(const float* __restrict__ acc, float* __restrict__ out)
{
  float total = 0.0f;
  for (int b = 0; b < Bv; ++b) {
    float vs = 0.0f, isum = 0.0f;
    for (int k = 0; k < Kv; ++k) {
      float vd = acc[OFF_VALID + b*Kv + k];
      vs += vd;
      if (vd > 0.0f) isum += acc[OFF_INST + b*Kv + k];
    }
    float obj = fmaxf(vs, 1.0f);
    float lb = (isum / obj)                                  // W_INST = 1
             + 10.0f * acc[OFF_VAR + b]                      // W_VAR  = 10 (already /obj)
             + (acc[OFF_BG + b] + acc[OFF_FG + b]) / (float)Nv; // W_SEED = FG_W = 1
    total += lb;
  }
  out[0] = total * (1.0f / Bv);
}

// ---------------------------------------------------------------------------
extern "C" void kernel_launch(void* const* d_in, const int* in_sizes, int n_in,
                              void* d_out, int out_size, void* d_ws, size_t ws_size,
                              hipStream_t stream)
{
  (void)in_sizes; (void)n_in; (void)out_size; (void)ws_size;
  const float* pred = (const float*)d_in[0];
  // d_in[1] = xyzm (recomputed analytically on-GPU; linspace is cheaper than HBM)
  const int* inst = (const int*)d_in[2];
  const int* lab  = (const int*)d_in[3];
  const unsigned char* cim = (const unsigned char*)d_in[4];  // jax bool -> 1 byte
  float* out = (float*)d_out;

  char* ws = (char*)d_ws;
  float*    acc   = (float*)ws;
  unsigned* keysA = (unsigned*)(ws + 4096);                               // 8*N u32
  unsigned* keysB = (unsigned*)(ws + 4096 + (size_t)Kv*Nv*4);             // N u32 pong
  unsigned* hist  = (unsigned*)(ws + 4096 + (size_t)Kv*Nv*4 + (size_t)Nv*4);
  unsigned* pos   = (unsigned*)((char*)hist + (size_t)256*NBLK*4);

  kInit<<<1, ACC_FLOATS, 0, stream>>>(acc);
  kStats<<<dim3(Nv/2048, Bv), 256, 0, stream>>>(pred, inst, lab, cim, acc);

  for (int b = 0; b < Bv; ++b) {
    kCoeff<<<1, 128, 0, stream>>>(acc, b);
    kDist<<<(Nv/16)/8, 256, 0, stream>>>(pred, inst, acc, keysA, b);

    for (int k = 0; k < Kv; ++k) {
      unsigned* s0 = keysA + (size_t)k * Nv;
      unsigned* bufs[2] = { s0, keysB };
      for (int pass = 0; pass < 4; ++pass) {       // 4x8-bit LSD, ends in s0
        const unsigned* src = bufs[pass & 1];
        unsigned*       dst = bufs[(pass + 1) & 1];
        int shift = pass * 8;
        kHist   <<<NBLK, 256, 0, stream>>>(src, hist, shift);
        kScanHist<<<1,   256, 0, stream>>>(hist);
        kScatter<<<NBLK, 256, 256*MAT_STRIDE*sizeof(unsigned), stream>>>(src, dst, hist, shift);
      }
      kPosCount<<<NBLK, 256, 0, stream>>>(s0, pos);
      kScanPos <<<1, 1, 0, stream>>>(pos);
      kLovasz  <<<NBLK, 256, 0, stream>>>(s0, pos, acc, b*Kv + k);
    }
  }
  kFinal<<<1, 1, 0, stream>>>(acc, out);
}